// EMAVectorQuantizer_59528246722962
// MI455X (gfx1250) — compile-verified
//
#include <hip/hip_runtime.h>

// ---------------- problem constants ----------------
#define NUM_CODES   8192      // codebook entries K
#define DIM         64        // codebook dim D
#define NTOK        32768     // B*H*W tokens
#define BATCH       32
#define HW          1024      // H*W
#define KD          (NUM_CODES * DIM)
#define ND          (NTOK * DIM)
#define DECAY       0.99f
#define OMD         0.01f     // 1 - DECAY
#define EPSV        1e-5f
#define BETA        0.25f

#define CHUNK       16        // codes per k-step
#define NCHUNK      (NUM_CODES / CHUNK)   // 512

typedef __attribute__((ext_vector_type(2))) float v2f;
typedef __attribute__((ext_vector_type(8))) float v8f;

// ---------------- workspace layout (float units) ----------------
#define WS_ZF      0                         // [ND]   z_flat, row-major [N,64]
#define WS_ENORM   (WS_ZF + ND)              // [K]    ||e||^2
#define WS_EMAACC  (WS_ENORM + NUM_CODES)    // [K*D]  segment sum of z (zeroed)
#define WS_NEWCL   (WS_EMAACC + KD)          // [K]    segment counts (zeroed)
#define WS_CS      (WS_NEWCL + NUM_CODES)    // [K]    EMA cluster size
#define WS_CSSUM   (WS_CS + NUM_CODES)       // [1]    sum of cs (zeroed)
#define WS_IDXI    (WS_CSSUM + 4)            // [N]    argmin indices (int32)

// ---------------- d_out layout (float units, concat of tuple) ----------------
#define O_ZQ    0                              // [B,C,H,W] = 2097152
#define O_LOSS  (O_ZQ + ND)                    // [1]
#define O_IDX   (O_LOSS + 1)                   // [N] as float
#define O_CSN   (O_IDX + NTOK)                 // [K]
#define O_EMAW  (O_CSN + NUM_CODES)            // [K*D]
#define O_EMB   (O_EMAW + KD)                  // [K*D]

// ---- CDNA5 async global->LDS copy (16B per lane), tracked by ASYNCcnt ----
__device__ __forceinline__ void async_copy_b128(void* lds_dst, const float* gsrc) {
  unsigned lds_addr = (unsigned)(uintptr_t)lds_dst;             // LDS aperture: low 32 bits = LDS byte offset
  unsigned long long gaddr = (unsigned long long)(uintptr_t)gsrc;
  asm volatile("global_load_async_to_lds_b128 %0, %1, off"
               :: "v"(lds_addr), "v"(gaddr)
               : "memory");
}

// ============================================================
// 1) NCHW -> [N, D] transpose via LDS tiles (coalesced in/out)
//    grid: (HW/32, DIM/32, BATCH), block: (32, 8)
// ============================================================
__global__ __launch_bounds__(256) void k_transpose(const float* __restrict__ z,
                                                   float* __restrict__ zf) {
  __shared__ float tile[32][33];
  const int b  = blockIdx.z;
  const int d0 = blockIdx.y * 32;
  const int p0 = blockIdx.x * 32;
  const int tx = threadIdx.x, ty = threadIdx.y;
#pragma unroll
  for (int i = 0; i < 32; i += 8) {
    tile[ty + i][tx] = z[(b * DIM + d0 + ty + i) * HW + p0 + tx];
  }
  __syncthreads();
#pragma unroll
  for (int i = 0; i < 32; i += 8) {
    zf[(b * HW + p0 + ty + i) * DIM + d0 + tx] = tile[tx][ty + i];
  }
}

// ============================================================
// 2) per-code squared norm
// ============================================================
__global__ __launch_bounds__(256) void k_enorm(const float* __restrict__ emb,
                                               float* __restrict__ enorm) {
  const int k = blockIdx.x * 256 + threadIdx.x;
  const float4* row = (const float4*)(emb + (size_t)k * DIM);
  float s = 0.f;
#pragma unroll
  for (int i = 0; i < DIM / 4; ++i) {
    float4 v = row[i];
    s += v.x * v.x + v.y * v.y + v.z * v.z + v.w * v.w;
  }
  enorm[k] = s;
}

// ============================================================
// 3) WMMA fp32 GEMM + streaming argmin.
//    Block: 8 waves, each wave owns TWO 16-token M-tiles (32 tokens)
//    -> 256 tokens/block, 128 blocks.
//    Codebook chunk (16 codes x 64 = 4KB) staged once per block in LDS
//    via async global->LDS DMA, double-buffered (ASYNCcnt + barriers).
//    score(n,k) = ||e_k||^2 - 2 * <z_n, e_k>   (||z||^2 dropped)
// ============================================================
__global__ __launch_bounds__(256) void k_argmin(const float* __restrict__ zf,
                                                const float* __restrict__ emb,
                                                const float* __restrict__ enorm,
                                                int* __restrict__ idx_i,
                                                float* __restrict__ idx_f) {
  __shared__ float ldsB[2][CHUNK * DIM];   // 2 x 4KB double buffer

  const int tid  = threadIdx.x;
  const int lane = tid & 31;
  const int wave = tid >> 5;
  const int col  = lane & 15;   // N / M position within 16
  const int hi   = lane >> 4;   // which K-pair of the 16x4 fragment
  const int m0   = (blockIdx.x * 8 + wave) * 32;   // first token of this wave

  // A fragments for both tiles, resident for the whole k-loop.
  // ISA 32-bit A 16x4 layout: lanes [0,16): K={0,1}; lanes [16,32): K={2,3}.
  v2f areg0[16], areg1[16];
  {
    const float* zr0 = zf + (size_t)(m0 + col) * DIM + 2 * hi;
    const float* zr1 = zf + (size_t)(m0 + 16 + col) * DIM + 2 * hi;
#pragma unroll
    for (int s = 0; s < 16; ++s) {
      areg0[s] = *(const v2f*)(zr0 + 4 * s);
      areg1[s] = *(const v2f*)(zr1 + 4 * s);
    }
  }

  float minv[16];
  int   mini[16];
#pragma unroll
  for (int j = 0; j < 16; ++j) { minv[j] = 3.0e38f; mini[j] = 0; }

  // per-thread staging slot: 16B of the 4KB chunk
  const int   srow = tid >> 4;          // 0..15 : code row within chunk
  const int   scol = (tid & 15) * 4;    // 0..60 : float column
  const float* gstage = emb + (size_t)srow * DIM + scol;   // + kb*DIM per chunk

  // prologue: prefetch chunks 0 and 1
  async_copy_b128(&ldsB[0][srow * DIM + scol], gstage + 0 * CHUNK * DIM);
  async_copy_b128(&ldsB[1][srow * DIM + scol], gstage + 1 * CHUNK * DIM);

  for (int c = 0; c < NCHUNK; ++c) {
    const int kb = c * CHUNK;
    // wait for this wave's oldest async (current buffer); keep prefetch in flight
    if (c + 1 < NCHUNK) asm volatile("s_wait_asynccnt 0x1" ::: "memory");
    else                asm volatile("s_wait_asynccnt 0x0" ::: "memory");
    __syncthreads();                         // all waves' chunk-c DMA visible

    const float* bbase = &ldsB[c & 1][col * DIM + 2 * hi];
    const float  en    = enorm[kb + col];

    // 4 independent accumulator chains (2 per tile); each LDS B-frag feeds 2 WMMAs
    v8f a0 = {}, a1 = {}, a2 = {}, a3 = {};
#pragma unroll
    for (int s = 0; s < 8; ++s) {
      v2f b0 = *(const v2f*)(bbase + 4 * (2 * s));       // ds_load_b64
      v2f b1 = *(const v2f*)(bbase + 4 * (2 * s + 1));   // ds_load_b64
      a0 = __builtin_amdgcn_wmma_f32_16x16x4_f32(false, areg0[2 * s],     false, b0, (short)0, a0, false, false);
      a2 = __builtin_amdgcn_wmma_f32_16x16x4_f32(false, areg1[2 * s],     false, b0, (short)0, a2, false, false);
      a1 = __builtin_amdgcn_wmma_f32_16x16x4_f32(false, areg0[2 * s + 1], false, b1, (short)0, a1, false, false);
      a3 = __builtin_amdgcn_wmma_f32_16x16x4_f32(false, areg1[2 * s + 1], false, b1, (short)0, a3, false, false);
    }
    v8f S0 = a0 + a1;   // tile0: tokens m0 + j + 8*hi
    v8f S1 = a2 + a3;   // tile1: tokens m0 + 16 + j + 8*hi

    const int code = kb + col;
#pragma unroll
    for (int j = 0; j < 8; ++j) {
      float sc0 = en - 2.0f * S0[j];
      if (sc0 < minv[j])     { minv[j]     = sc0; mini[j]     = code; }
      float sc1 = en - 2.0f * S1[j];
      if (sc1 < minv[8 + j]) { minv[8 + j] = sc1; mini[8 + j] = code; }
    }

    __syncthreads();                         // everyone done reading buf[c&1]
    if (c + 2 < NCHUNK)                      // refill it with chunk c+2
      async_copy_b128(&ldsB[c & 1][srow * DIM + scol],
                      gstage + (size_t)(kb + 2 * CHUNK) * DIM);
  }

  // reduce across the 16 lanes of each half-wave; tie-break toward lower index
#pragma unroll
  for (int j = 0; j < 16; ++j) {
    float v = minv[j]; int id = mini[j];
#pragma unroll
    for (int off = 8; off >= 1; off >>= 1) {
      float ov = __shfl_xor(v, off, 32);
      int   oi = __shfl_xor(id, off, 32);
      if (ov < v || (ov == v && oi < id)) { v = ov; id = oi; }
    }
    minv[j] = v; mini[j] = id;
  }
  if (col == 0) {
#pragma unroll
    for (int j = 0; j < 8; ++j) {
      const int t0 = m0 + j + 8 * hi;        // tile 0
      idx_i[t0] = mini[j];
      idx_f[t0] = (float)mini[j];
      const int t1 = m0 + 16 + j + 8 * hi;   // tile 1
      idx_i[t1] = mini[8 + j];
      idx_f[t1] = (float)mini[8 + j];
    }
  }
}

// ============================================================
// 4) segment sums via atomics: counts + per-code z sums
// ============================================================
__global__ __launch_bounds__(256) void k_scatter(const float* __restrict__ zf,
                                                 const int* __restrict__ idx_i,
                                                 float* __restrict__ ema_acc,
                                                 float* __restrict__ new_cluster) {
  const int id = blockIdx.x * 256 + threadIdx.x;   // id = n*64 + d
  const int n = id >> 6, d = id & 63;
  const int i = idx_i[n];
  atomicAdd(&ema_acc[(size_t)i * DIM + d], zf[id]);
  if (d == 0) atomicAdd(&new_cluster[i], 1.0f);
}

// ============================================================
// 5) EMA cluster size + global sum
// ============================================================
__global__ __launch_bounds__(256) void k_cs(const float* __restrict__ cluster_size,
                                            const float* __restrict__ new_cluster,
                                            float* __restrict__ cs,
                                            float* __restrict__ cs_sum) {
  __shared__ float red[256];
  const int k = blockIdx.x * 256 + threadIdx.x;
  const float c = DECAY * cluster_size[k] + OMD * new_cluster[k];
  cs[k] = c;
  red[threadIdx.x] = c;
  __syncthreads();
  for (int s = 128; s > 0; s >>= 1) {
    if (threadIdx.x < s) red[threadIdx.x] += red[threadIdx.x + s];
    __syncthreads();
  }
  if (threadIdx.x == 0) atomicAdd(cs_sum, red[0]);
}

// ============================================================
// 6) EMA weights
// ============================================================
__global__ __launch_bounds__(256) void k_emaw(const float* __restrict__ ema_w,
                                              const float* __restrict__ ema_acc,
                                              float* __restrict__ ema_w_out) {
  const int id = blockIdx.x * 256 + threadIdx.x;
  ema_w_out[id] = DECAY * ema_w[id] + OMD * ema_acc[id];
}

// ============================================================
// 7) cs_norm + normalized codebook
// ============================================================
__global__ __launch_bounds__(256) void k_embout(const float* __restrict__ ema_w_out,
                                                const float* __restrict__ cs,
                                                const float* __restrict__ cs_sum,
                                                float* __restrict__ cs_norm_out,
                                                float* __restrict__ emb_out) {
  const int id = blockIdx.x * 256 + threadIdx.x;
  const int k = id >> 6;
  const float denom = cs_sum[0] + (float)NUM_CODES * EPSV;
  const float csn = (cs[k] + EPSV) / denom;
  emb_out[id] = ema_w_out[id] / csn;
  if ((id & 63) == 0) cs_norm_out[k] = csn;
}

// ============================================================
// 8) z_q gather -> NCHW output (z_q_st == z_q numerically) + loss
//    id = (b*64 + d)*1024 + p  -> coalesced output writes
// ============================================================
__global__ __launch_bounds__(256) void k_output(const float* __restrict__ zf,
                                                const float* __restrict__ emb,
                                                const int* __restrict__ idx_i,
                                                float* __restrict__ zq_out,
                                                float* __restrict__ loss_acc) {
  __shared__ float red[256];
  const int id = blockIdx.x * 256 + threadIdx.x;
  const int p  = id & (HW - 1);
  const int bd = id >> 10;
  const int d  = bd & 63;
  const int b  = bd >> 6;
  const int n  = b * HW + p;
  const int i  = idx_i[n];
  const float q  = emb[(size_t)i * DIM + d];
  const float zv = zf[(size_t)n * DIM + d];
  const float diff = q - zv;
  zq_out[id] = q;
  red[threadIdx.x] = diff * diff;
  __syncthreads();
  for (int s = 128; s > 0; s >>= 1) {
    if (threadIdx.x < s) red[threadIdx.x] += red[threadIdx.x + s];
    __syncthreads();
  }
  if (threadIdx.x == 0)
    atomicAdd(loss_acc, red[0] * (BETA / (float)ND));
}

// ============================================================
extern "C" void kernel_launch(void* const* d_in, const int* in_sizes, int n_in,
                              void* d_out, int out_size, void* d_ws, size_t ws_size,
                              hipStream_t stream) {
  const float* z            = (const float*)d_in[0];
  const float* emb          = (const float*)d_in[1];
  const float* ema_w        = (const float*)d_in[2];
  const float* cluster_size = (const float*)d_in[3];
  float* out = (float*)d_out;
  float* ws  = (float*)d_ws;

  float* zf       = ws + WS_ZF;
  float* enorm    = ws + WS_ENORM;
  float* ema_acc  = ws + WS_EMAACC;
  float* new_cl   = ws + WS_NEWCL;
  float* cs       = ws + WS_CS;
  float* cs_sum   = ws + WS_CSSUM;
  int*   idx_i    = (int*)(ws + WS_IDXI);

  // zero accumulators (graph-capture-safe async memsets)
  hipMemsetAsync(ema_acc, 0, (size_t)(KD + NUM_CODES) * sizeof(float), stream); // ema_acc + new_cluster
  hipMemsetAsync(cs_sum, 0, sizeof(float), stream);
  hipMemsetAsync(out + O_LOSS, 0, sizeof(float), stream);

  k_transpose<<<dim3(HW / 32, DIM / 32, BATCH), dim3(32, 8), 0, stream>>>(z, zf);
  k_enorm<<<NUM_CODES / 256, 256, 0, stream>>>(emb, enorm);
  k_argmin<<<NTOK / 256, 256, 0, stream>>>(zf, emb, enorm, idx_i, out + O_IDX);
  k_scatter<<<ND / 256, 256, 0, stream>>>(zf, idx_i, ema_acc, new_cl);
  k_cs<<<NUM_CODES / 256, 256, 0, stream>>>(cluster_size, new_cl, cs, cs_sum);
  k_emaw<<<KD / 256, 256, 0, stream>>>(ema_w, ema_acc, out + O_EMAW);
  k_embout<<<KD / 256, 256, 0, stream>>>(out + O_EMAW, cs, cs_sum, out + O_CSN, out + O_EMB);
  k_output<<<ND / 256, 256, 0, stream>>>(zf, emb, idx_i, out + O_ZQ, out + O_LOSS);
}